// NegF1_77129022701690
// MI455X (gfx1250) — compile-verified
//
#include <hip/hip_runtime.h>

typedef float v2f __attribute__((ext_vector_type(2)));
typedef float v4f __attribute__((ext_vector_type(4)));
typedef int   v4i __attribute__((ext_vector_type(4)));
typedef float v8f __attribute__((ext_vector_type(8)));

#define NBLK 2048
#define NTHR 256
#define NWAVE (NTHR / 32)
#define F1_EPS 1e-5f

// Per-element soft confusion-matrix contributions.
__device__ __forceinline__ void contrib(float p, int l,
                                        float& tp, float& fp, float& fn) {
    bool pos_pred = p > 0.5f;
    bool pos_lbl  = (l == 1);
    tp += (pos_pred &&  pos_lbl) ? p          : 0.0f;
    fp += (pos_pred && !pos_lbl) ? p          : 0.0f;
    fn += (!pos_pred &&  pos_lbl) ? (1.0f - p) : 0.0f;
}

// Accumulate a 2-element A column-pair into C via the matrix pipe:
// D = A(16x4) * ones(4x16) + C  -> every row of C accumulates the row-sum of A.
__device__ __forceinline__ v8f wmma_acc(float a0, float a1, v2f ones, v8f c) {
    v2f a;
    a.x = a0;
    a.y = a1;
    return __builtin_amdgcn_wmma_f32_16x16x4_f32(
        /*neg_a=*/false, a, /*neg_b=*/false, ones,
        /*c_mod=*/(short)0, c, /*reuse_a=*/false, /*reuse_b=*/false);
}

__global__ void __launch_bounds__(NTHR)
f1_partial_kernel(const float* __restrict__ probs,
                  const int* __restrict__ lbls,
                  float* __restrict__ partials,
                  long long n) {
    const long long tid      = (long long)blockIdx.x * blockDim.x + threadIdx.x;
    const long long nthreads = (long long)gridDim.x * blockDim.x;

    v8f c_tp = {};
    v8f c_fp = {};
    v8f c_fn = {};
    v2f ones;
    ones.x = 1.0f;
    ones.y = 1.0f;

    const v4f* __restrict__ p4 = (const v4f*)probs;
    const v4i* __restrict__ l4 = (const v4i*)lbls;

    // Main loop: 8 elements per lane per iteration (two b128 prob loads + two
    // b128 label loads), accumulated on the matrix pipe (3 WMMAs / iter).
    const long long n8 = n >> 3;
    for (long long g = tid; g < n8; g += nthreads) {
        v4f pa = p4[2 * g + 0];
        v4f pb = p4[2 * g + 1];
        v4i la = l4[2 * g + 0];
        v4i lb = l4[2 * g + 1];

        float tp0 = 0.0f, fp0 = 0.0f, fn0 = 0.0f;
        float tp1 = 0.0f, fp1 = 0.0f, fn1 = 0.0f;
        contrib(pa.x, la.x, tp0, fp0, fn0);
        contrib(pa.y, la.y, tp0, fp0, fn0);
        contrib(pa.z, la.z, tp0, fp0, fn0);
        contrib(pa.w, la.w, tp0, fp0, fn0);
        contrib(pb.x, lb.x, tp1, fp1, fn1);
        contrib(pb.y, lb.y, tp1, fp1, fn1);
        contrib(pb.z, lb.z, tp1, fp1, fn1);
        contrib(pb.w, lb.w, tp1, fp1, fn1);

        c_tp = wmma_acc(tp0, tp1, ones, c_tp);
        c_fp = wmma_acc(fp0, fp1, ones, c_fp);
        c_fn = wmma_acc(fn0, fn1, ones, c_fn);
    }

    // Scalar tail (n % 8 leftovers), folded in through one WMMA per quantity
    // so all WMMAs execute with full EXEC.
    float ttp = 0.0f, tfp = 0.0f, tfn = 0.0f;
    for (long long i = (n8 << 3) + tid; i < n; i += nthreads) {
        contrib(probs[i], lbls[i], ttp, tfp, tfn);
    }
    c_tp = wmma_acc(ttp, 0.0f, ones, c_tp);
    c_fp = wmma_acc(tfp, 0.0f, ones, c_fp);
    c_fn = wmma_acc(tfn, 0.0f, ones, c_fn);

    // C layout (16x16 f32): lanes 0-15 hold rows M=0..7 (VGPR v -> row v),
    // lanes 16-31 hold rows M=8..15, column = lane%16, and every column of a
    // row carries the identical accumulated row-sum. Summing a lane's 8 C
    // elements gives the half-wave total; xor-16 shuffle completes the wave.
    float s_tp = 0.0f, s_fp = 0.0f, s_fn = 0.0f;
#pragma unroll
    for (int i = 0; i < 8; ++i) {
        s_tp += c_tp[i];
        s_fp += c_fp[i];
        s_fn += c_fn[i];
    }
    s_tp += __shfl_xor(s_tp, 16, 32);
    s_fp += __shfl_xor(s_fp, 16, 32);
    s_fn += __shfl_xor(s_fn, 16, 32);

    // Block reduction across the 8 waves.
    __shared__ float red[3][NWAVE];
    const int lane = threadIdx.x & 31;
    const int wave = threadIdx.x >> 5;
    if (lane == 0) {
        red[0][wave] = s_tp;
        red[1][wave] = s_fp;
        red[2][wave] = s_fn;
    }
    __syncthreads();
    if (threadIdx.x == 0) {
        float btp = 0.0f, bfp = 0.0f, bfn = 0.0f;
#pragma unroll
        for (int w = 0; w < NWAVE; ++w) {
            btp += red[0][w];
            bfp += red[1][w];
            bfn += red[2][w];
        }
        partials[3 * blockIdx.x + 0] = btp;
        partials[3 * blockIdx.x + 1] = bfp;
        partials[3 * blockIdx.x + 2] = bfn;
    }
}

__global__ void __launch_bounds__(256)
f1_final_kernel(const float* __restrict__ partials, int nblk,
                float* __restrict__ out) {
    float t = 0.0f, f = 0.0f, g = 0.0f;
    for (int i = threadIdx.x; i < nblk; i += blockDim.x) {
        t += partials[3 * i + 0];
        f += partials[3 * i + 1];
        g += partials[3 * i + 2];
    }
#pragma unroll
    for (int off = 16; off > 0; off >>= 1) {
        t += __shfl_down(t, off, 32);
        f += __shfl_down(f, off, 32);
        g += __shfl_down(g, off, 32);
    }
    __shared__ float red[3][8];
    const int lane = threadIdx.x & 31;
    const int wave = threadIdx.x >> 5;
    if (lane == 0) {
        red[0][wave] = t;
        red[1][wave] = f;
        red[2][wave] = g;
    }
    __syncthreads();
    if (threadIdx.x == 0) {
        float TP = 0.0f, FP = 0.0f, FN = 0.0f;
#pragma unroll
        for (int w = 0; w < 8; ++w) {
            TP += red[0][w];
            FP += red[1][w];
            FN += red[2][w];
        }
        float precision = (TP + F1_EPS) / (TP + FP + F1_EPS);
        float recall    = (TP + F1_EPS) / (TP + FN + F1_EPS);
        float f1        = 2.0f * precision * recall / (precision + recall);
        out[0] = -f1;
    }
}

extern "C" void kernel_launch(void* const* d_in, const int* in_sizes, int n_in,
                              void* d_out, int out_size, void* d_ws, size_t ws_size,
                              hipStream_t stream) {
    const float* probs = (const float*)d_in[0];
    const int*   lbls  = (const int*)d_in[1];
    const long long n  = (long long)in_sizes[0];

    float* partials = (float*)d_ws;

    int nblk = NBLK;
    // Safety clamp: never write more partials than the scratch buffer holds.
    size_t need = (size_t)nblk * 3u * sizeof(float);
    if (ws_size < need) {
        nblk = (int)(ws_size / (3u * sizeof(float)));
        if (nblk < 1) nblk = 1;
    }

    f1_partial_kernel<<<nblk, NTHR, 0, stream>>>(probs, lbls, partials, n);
    f1_final_kernel<<<1, 256, 0, stream>>>(partials, nblk, (float*)d_out);
}